// UpdateONNX_77730318123549
// MI455X (gfx1250) — compile-verified
//
#include <hip/hip_runtime.h>
#include <math.h>

// ---------------------------------------------------------------------------
// Problem constants
// ---------------------------------------------------------------------------
#define DIMC   384
#define NTOK   4096
#define CCRAW  882
#define CCPAD  896          // 882 padded to multiple of BK(=32)

#define BM 64
#define BN 128
#define BK 32
#define LDP 34              // LDS row pitch (words): even -> 8B-aligned b64 pairs,
                            // 34*n mod 64 distinct over n=0..15 -> conflict-free frags

typedef float v2f __attribute__((ext_vector_type(2)));
typedef float v8f __attribute__((ext_vector_type(8)));

// Epilogue modes
enum {
  EPI_NONE = 0,      // out = acc + bias
  EPI_RELU,          // out = max(acc+bias, 0)
  EPI_SIGMOID,       // out = sigmoid(acc+bias)
  EPI_EXP,           // out = exp(acc+bias)
  EPI_ADD,           // out = add0[m][n] + acc + bias
  EPI_ADD_CHMAJOR2,  // out = acc + bias + add0[n*M+m] + add1[n*M+m]
  EPI_GATED_ADD      // out = add0[m][n] + add1[m][n] * (acc + bias)
};

// ---------------------------------------------------------------------------
// Generic WMMA fp32 GEMM: C[M x Nout] = epi(A[M x K] @ W[K x Nout] + bias)
// Optional row gather on A (A row used for output row m is gidx[m]).
// Block tile 64x128, 256 threads = 8 wave32 waves; each wave: 16(M) x 64(N)
// = 4 accumulators sharing one A fragment (5 ds_load_b64 per 4 WMMAs).
// Uses V_WMMA_F32_16X16X4_F32; fragments per ISA 7.12.2 32-bit layouts:
//   A frag: lane L: row = (L&15), k-pair at koff = 2*(L>>4)
//   B frag: lane L: col = (L&15), same k-pair mapping
//   C/D:    VGPR r holds M = r + 8*(L>>4), N = (L&15)
// ---------------------------------------------------------------------------
__global__ __launch_bounds__(256)
void gemm_wmma_f32(const float* __restrict__ A, int K,
                   const int* __restrict__ gidx,
                   const float* __restrict__ W,
                   const float* __restrict__ bias,
                   float* __restrict__ C,
                   const float* __restrict__ add0,
                   const float* __restrict__ add1,
                   int M, int Nout, int mode)
{
  __shared__ float As [BM][LDP];   // k-major  (A rows contiguous in k)
  __shared__ float BsT[BN][LDP];   // n-major  (B k-pairs contiguous -> b64 frags)

  const int tid  = threadIdx.x;
  const int wave = tid >> 5;         // 0..7
  const int lane = tid & 31;
  const int mw   = wave & 3;         // M sub-tile (16 rows)
  const int nw   = wave >> 2;        // N group (64 cols)

  const int bm = blockIdx.x * BM;
  const int bn = blockIdx.y * BN;

  v8f acc[4];
  #pragma unroll
  for (int t = 0; t < 4; ++t) acc[t] = (v8f){0.f,0.f,0.f,0.f,0.f,0.f,0.f,0.f};

  const int mrow = (mw << 4) + (lane & 15);    // A-frag row within block tile
  const int koff = (lane >> 4) << 1;           // 0 (lanes 0-15) / 2 (lanes 16-31)
  const int ncol = (nw << 6) + (lane & 15);    // B-frag base column

  for (int k0 = 0; k0 < K; k0 += BK) {
    __syncthreads();
    // ---- stage A tile: 64 rows x 32 k (coalesced float4 along k) ----
    #pragma unroll
    for (int i = tid; i < BM * (BK / 4); i += 256) {
      int row = i >> 3;
      int kq  = (i & 7) << 2;
      int gm  = bm + row;
      int ar  = gidx ? gidx[gm] : gm;
      const float4 v = *(const float4*)(A + (size_t)ar * K + k0 + kq);
      *(v2f*)&As[row][kq + 0] = (v2f){v.x, v.y};
      *(v2f*)&As[row][kq + 2] = (v2f){v.z, v.w};
    }
    // ---- stage B tile transposed: 32 k x 128 n (coalesced float4 along n) ----
    #pragma unroll
    for (int i = tid; i < BK * (BN / 4); i += 256) {
      int kr = i >> 5;            // 0..31
      int nq = (i & 31) << 2;     // 0..124
      const float4 v = *(const float4*)(W + (size_t)(k0 + kr) * Nout + bn + nq);
      BsT[nq + 0][kr] = v.x;
      BsT[nq + 1][kr] = v.y;
      BsT[nq + 2][kr] = v.z;
      BsT[nq + 3][kr] = v.w;
    }
    __syncthreads();

    // ---- 8 WMMA K-steps of 4; one A frag feeds 4 N tiles ----
    #pragma unroll
    for (int ks = 0; ks < BK; ks += 4) {
      const v2f a = *(const v2f*)&As[mrow][ks + koff];
      #pragma unroll
      for (int t = 0; t < 4; ++t) {
        const v2f b = *(const v2f*)&BsT[ncol + (t << 4)][ks + koff];
        acc[t] = __builtin_amdgcn_wmma_f32_16x16x4_f32(false, a, false, b,
                                                       (short)0, acc[t],
                                                       false, false);
      }
    }
  }

  // ---- epilogue ----
  const int mbase = bm + (mw << 4) + ((lane >> 4) << 3);
  const int nbase = bn + (nw << 6) + (lane & 15);
  #pragma unroll
  for (int r = 0; r < 8; ++r) {
    const int m = mbase + r;
    #pragma unroll
    for (int t = 0; t < 4; ++t) {
      const int n = nbase + (t << 4);
      float v = acc[t][r] + bias[n];
      float out;
      switch (mode) {
        default:
        case EPI_NONE:    out = v; break;
        case EPI_RELU:    out = fmaxf(v, 0.f); break;
        case EPI_SIGMOID: out = 1.f / (1.f + expf(-v)); break;
        case EPI_EXP:     out = expf(v); break;
        case EPI_ADD:     out = add0[(size_t)m * Nout + n] + v; break;
        case EPI_ADD_CHMAJOR2:
          out = v + add0[(size_t)n * M + m] + add1[(size_t)n * M + m];
          break;
        case EPI_GATED_ADD:
          out = add0[(size_t)m * Nout + n] + add1[(size_t)m * Nout + n] * v;
          break;
      }
      C[(size_t)m * Nout + n] = out;
    }
  }
}

// ---------------------------------------------------------------------------
// corr (1,882,4096,1) channel-major -> corrT [4096][896] token-major, zero pad
// ---------------------------------------------------------------------------
__global__ __launch_bounds__(256)
void transpose_pad_kernel(const float* __restrict__ corr, float* __restrict__ corrT)
{
  __shared__ float tile[32][33];
  const int tb = blockIdx.x * 32;   // token block
  const int cb = blockIdx.y * 32;   // channel block
  const int x = threadIdx.x;        // 0..31
  const int y = threadIdx.y;        // 0..7
  #pragma unroll
  for (int yy = y; yy < 32; yy += 8) {
    int ch = cb + yy;
    tile[yy][x] = (ch < CCRAW) ? corr[(size_t)ch * NTOK + tb + x] : 0.f;
  }
  __syncthreads();
  #pragma unroll
  for (int yy = y; yy < 32; yy += 8) {
    corrT[(size_t)(tb + yy) * CCPAD + cb + x] = tile[x][yy];
  }
}

// pad corr0_w [882][384] -> [896][384] (zero rows 882..895)
__global__ void pad_w0_kernel(const float* __restrict__ w, float* __restrict__ wp)
{
  size_t i = (size_t)blockIdx.x * blockDim.x + threadIdx.x;
  if (i >= (size_t)CCPAD * DIMC) return;
  int k = (int)(i / DIMC);
  int n = (int)(i % DIMC);
  wp[i] = (k < CCRAW) ? w[(size_t)k * DIMC + n] : 0.f;
}

// ---------------------------------------------------------------------------
// LayerNorm over last dim (384), one 128-thread block per row
// ---------------------------------------------------------------------------
__global__ __launch_bounds__(128)
void layernorm_kernel(const float* __restrict__ x,
                      const float* __restrict__ g,
                      const float* __restrict__ b,
                      float* __restrict__ y, int do_relu)
{
  const int row = blockIdx.x;
  const int t = threadIdx.x;
  const float* xr = x + (size_t)row * DIMC;
  float v0 = xr[t], v1 = xr[t + 128], v2 = xr[t + 256];
  __shared__ float rs[128], rss[128];
  rs[t]  = v0 + v1 + v2;
  rss[t] = v0 * v0 + v1 * v1 + v2 * v2;
  __syncthreads();
  for (int o = 64; o > 0; o >>= 1) {
    if (t < o) { rs[t] += rs[t + o]; rss[t] += rss[t + o]; }
    __syncthreads();
  }
  const float mean = rs[0] * (1.f / DIMC);
  const float var  = rss[0] * (1.f / DIMC) - mean * mean;
  const float inv  = rsqrtf(var + 1e-3f);
  float* yr = y + (size_t)row * DIMC;
  #pragma unroll
  for (int q = 0; q < 3; ++q) {
    int c = t + q * 128;
    float v = (q == 0 ? v0 : (q == 1 ? v1 : v2));
    float o = (v - mean) * inv * g[c] + b[c];
    yr[c] = do_relu ? fmaxf(o, 0.f) : o;
  }
}

// ---------------------------------------------------------------------------
// Neighbor search: exact argmax/argmin semantics of the reference
// ---------------------------------------------------------------------------
__global__ __launch_bounds__(256)
void neighbors_kernel(const int* __restrict__ kk, const int* __restrict__ jj,
                      int* __restrict__ ix, int* __restrict__ jx)
{
  __shared__ int skk[NTOK];
  __shared__ int sjj[NTOK];
  for (int i = threadIdx.x; i < NTOK; i += 256) { skk[i] = kk[i]; sjj[i] = jj[i]; }
  __syncthreads();
  const int t = blockIdx.x * 256 + threadIdx.x;
  const int kt = skk[t], jt = sjj[t];
  int bv = (skk[0] == kt && sjj[0] < jt) ? sjj[0] : 0;      int bi = 0;
  int cv = (skk[0] == kt && sjj[0] > jt) ? sjj[0] : NTOK;   int ci = 0;
  for (int s = 1; s < NTOK; ++s) {
    const int js = sjj[s];
    const bool m = (skk[s] == kt);
    const int pv = (m && js < jt) ? js : 0;
    if (pv > bv) { bv = pv; bi = s; }
    const int nv = (m && js > jt) ? js : NTOK;
    if (nv < cv) { cv = nv; ci = s; }
  }
  ix[t] = bi;
  jx[t] = ci;
}

__global__ void make_idx_kernel(const int* __restrict__ ii, const int* __restrict__ jj,
                                const int* __restrict__ kk,
                                int* __restrict__ ikk, int* __restrict__ iij)
{
  int t = blockIdx.x * blockDim.x + threadIdx.x;
  if (t >= NTOK) return;
  int a = kk[t];               ikk[t] = min(max(a, 0), NTOK - 1);
  long long b = (long long)ii[t] * 12345 + jj[t];
  iij[t] = (int)min((long long)(NTOK - 1), max((long long)0, b));
}

__global__ void zero_kernel(float* __restrict__ p, size_t n)
{
  size_t i = (size_t)blockIdx.x * blockDim.x + threadIdx.x;
  if (i < n) p[i] = 0.f;
}

// dst[idx[t]][c] += src[t][c]
__global__ void scatter_add_kernel(const float* __restrict__ src,
                                   const int* __restrict__ idx,
                                   float* __restrict__ dst)
{
  size_t i = (size_t)blockIdx.x * blockDim.x + threadIdx.x;
  if (i >= (size_t)NTOK * DIMC) return;
  int t = (int)(i / DIMC);
  int c = (int)(i % DIMC);
  atomicAdd(&dst[(size_t)idx[t] * DIMC + c], src[i]);
}

// out = f * e / max(denom, 1e-6)
__global__ void weighted_kernel(const float* __restrict__ e,
                                const float* __restrict__ denom,
                                const float* __restrict__ f,
                                float* __restrict__ out)
{
  size_t i = (size_t)blockIdx.x * blockDim.x + threadIdx.x;
  if (i >= (size_t)NTOK * DIMC) return;
  out[i] = f[i] * (e[i] / fmaxf(denom[i], 1e-6f));
}

// ---------------------------------------------------------------------------
// Output heads: d = relu(net)@d_w + d_b ; w = sigmoid(relu(net)@w_w + w_b)
// One wave32 per token, shuffle reduction.
// ---------------------------------------------------------------------------
__global__ __launch_bounds__(256)
void heads_kernel(const float* __restrict__ net,
                  const float* __restrict__ dw, const float* __restrict__ db,
                  const float* __restrict__ ww, const float* __restrict__ wb,
                  float* __restrict__ dout, float* __restrict__ wout)
{
  const int t = blockIdx.x * 8 + (threadIdx.x >> 5);
  const int lane = threadIdx.x & 31;
  const float* xr = net + (size_t)t * DIMC;
  float a0 = 0.f, a1 = 0.f, a2 = 0.f, a3 = 0.f;
  for (int c = lane; c < DIMC; c += 32) {
    float r = fmaxf(xr[c], 0.f);
    a0 += r * dw[c * 2 + 0];
    a1 += r * dw[c * 2 + 1];
    a2 += r * ww[c * 2 + 0];
    a3 += r * ww[c * 2 + 1];
  }
  #pragma unroll
  for (int o = 16; o > 0; o >>= 1) {
    a0 += __shfl_down(a0, o, 32);
    a1 += __shfl_down(a1, o, 32);
    a2 += __shfl_down(a2, o, 32);
    a3 += __shfl_down(a3, o, 32);
  }
  if (lane == 0) {
    dout[t * 2 + 0] = a0 + db[0];
    dout[t * 2 + 1] = a1 + db[1];
    wout[t * 2 + 0] = 1.f / (1.f + expf(-(a2 + wb[0])));
    wout[t * 2 + 1] = 1.f / (1.f + expf(-(a3 + wb[1])));
  }
}

// ---------------------------------------------------------------------------
// Host orchestration
// ---------------------------------------------------------------------------
enum {
  IN_NET = 0, IN_INP, IN_CORR, IN_II, IN_JJ, IN_KK,
  P_CORR0_W, P_CORR0_B, P_CORR2_W, P_CORR2_B, P_CORR5_W, P_CORR5_B,
  P_CORR_LN_G, P_CORR_LN_B, P_NORM_G, P_NORM_B,
  P_C1A_W, P_C1A_B, P_C1B_W, P_C1B_B,
  P_C2A_W, P_C2A_B, P_C2B_W, P_C2B_B,
  P_AKK_F_W, P_AKK_F_B, P_AKK_G_W, P_AKK_G_B, P_AKK_H_W, P_AKK_H_B,
  P_AIJ_F_W, P_AIJ_F_B, P_AIJ_G_W, P_AIJ_G_B, P_AIJ_H_W, P_AIJ_H_B,
  P_GR1_GATE_W, P_GR1_GATE_B, P_GR1_RES1_W, P_GR1_RES1_B, P_GR1_RES2_W, P_GR1_RES2_B,
  P_GR2_GATE_W, P_GR2_GATE_B, P_GR2_RES1_W, P_GR2_RES1_B, P_GR2_RES2_W, P_GR2_RES2_B,
  P_GRU_LN1_G, P_GRU_LN1_B, P_GRU_LN2_G, P_GRU_LN2_B,
  P_D_W, P_D_B, P_W_W, P_W_B
};

static inline void launch_gemm(hipStream_t s, const float* A, int K, const int* gidx,
                               const float* W, const float* bias, float* C,
                               const float* a0, const float* a1, int mode)
{
  dim3 grid(NTOK / BM, DIMC / BN);
  gemm_wmma_f32<<<grid, 256, 0, s>>>(A, K, gidx, W, bias, C, a0, a1, NTOK, DIMC, mode);
}

extern "C" void kernel_launch(void* const* d_in, const int* in_sizes, int n_in,
                              void* d_out, int out_size, void* d_ws, size_t ws_size,
                              hipStream_t stream)
{
  (void)in_sizes; (void)n_in; (void)out_size; (void)ws_size;
  const float* f_in[64];
  for (int i = 0; i < 56; ++i) f_in[i] = (const float*)d_in[i];
  const int* jj = (const int*)d_in[IN_JJ];
  const int* kk = (const int*)d_in[IN_KK];

  float* ws = (float*)d_ws;
  const size_t SLOT = (size_t)NTOK * DIMC;            // 1,572,864 floats
  float* corrT = ws;                                  // [4096][896]
  float* denom = ws;                                  // overlays corrT (later)
  float* ybuf  = ws + SLOT;                           // overlays corrT (later)
  float* w0pad = ws + (size_t)NTOK * CCPAD;           // [896][384]
  float* buf1  = w0pad + (size_t)CCPAD * DIMC;
  float* buf2  = buf1 + SLOT;
  float* buf3  = buf2 + SLOT;
  float* netA  = buf3 + SLOT;
  float* netB  = netA + SLOT;
  int*   ix    = (int*)(netB + SLOT);
  int*   jx    = ix + NTOK;
  int*   ikk   = jx + NTOK;
  int*   iij   = ikk + NTOK;

  float* out_net = (float*)d_out;                     // [4096][384]
  float* out_d   = out_net + SLOT;                    // [4096][2]
  float* out_w   = out_d + (size_t)NTOK * 2;          // [4096][2]

  const int EW_BLOCKS = (int)((SLOT + 255) / 256);

  // --- stage 0: layout prep + indices ---
  transpose_pad_kernel<<<dim3(NTOK / 32, CCPAD / 32), dim3(32, 8), 0, stream>>>(
      f_in[IN_CORR], corrT);
  pad_w0_kernel<<<(int)(((size_t)CCPAD * DIMC + 255) / 256), 256, 0, stream>>>(
      f_in[P_CORR0_W], w0pad);
  neighbors_kernel<<<NTOK / 256, 256, 0, stream>>>(kk, jj, ix, jx);
  make_idx_kernel<<<NTOK / 256, 256, 0, stream>>>(
      (const int*)d_in[IN_II], jj, kk, ikk, iij);

  // --- corr MLP: c = Lin(relu(LN(Lin(relu(Lin(corr)))))) ---
  launch_gemm(stream, corrT, CCPAD, nullptr, w0pad, f_in[P_CORR0_B], buf1,
              nullptr, nullptr, EPI_RELU);
  launch_gemm(stream, buf1, DIMC, nullptr, f_in[P_CORR2_W], f_in[P_CORR2_B], buf2,
              nullptr, nullptr, EPI_NONE);
  layernorm_kernel<<<NTOK, 128, 0, stream>>>(buf2, f_in[P_CORR_LN_G], f_in[P_CORR_LN_B],
                                             buf1, 1);
  // net = net + inp + c, fused into corr5 epilogue (net/inp are channel-major)
  launch_gemm(stream, buf1, DIMC, nullptr, f_in[P_CORR5_W], f_in[P_CORR5_B], buf3,
              f_in[IN_NET], f_in[IN_INP], EPI_ADD_CHMAJOR2);
  layernorm_kernel<<<NTOK, 128, 0, stream>>>(buf3, f_in[P_NORM_G], f_in[P_NORM_B],
                                             netA, 0);

  // --- neighbor MLPs: net += mlp2(net[ix]); net += mlp2(net[jx]) ---
  launch_gemm(stream, netA, DIMC, ix, f_in[P_C1A_W], f_in[P_C1A_B], buf1,
              nullptr, nullptr, EPI_RELU);
  launch_gemm(stream, buf1, DIMC, nullptr, f_in[P_C1B_W], f_in[P_C1B_B], netB,
              netA, nullptr, EPI_ADD);
  launch_gemm(stream, netB, DIMC, jx, f_in[P_C2A_W], f_in[P_C2A_B], buf1,
              nullptr, nullptr, EPI_RELU);
  launch_gemm(stream, buf1, DIMC, nullptr, f_in[P_C2B_W], f_in[P_C2B_B], netA,
              netB, nullptr, EPI_ADD);

  // --- soft_agg(kk): netA -> netB ---
  launch_gemm(stream, netA, DIMC, nullptr, f_in[P_AKK_G_W], f_in[P_AKK_G_B], buf2,
              nullptr, nullptr, EPI_EXP);
  zero_kernel<<<2 * EW_BLOCKS, 256, 0, stream>>>(denom, 2 * SLOT);   // denom + ybuf
  scatter_add_kernel<<<EW_BLOCKS, 256, 0, stream>>>(buf2, ikk, denom);
  launch_gemm(stream, netA, DIMC, nullptr, f_in[P_AKK_F_W], f_in[P_AKK_F_B], buf1,
              nullptr, nullptr, EPI_NONE);
  weighted_kernel<<<EW_BLOCKS, 256, 0, stream>>>(buf2, denom, buf1, buf3);
  scatter_add_kernel<<<EW_BLOCKS, 256, 0, stream>>>(buf3, ikk, ybuf);
  launch_gemm(stream, ybuf, DIMC, nullptr, f_in[P_AKK_H_W], f_in[P_AKK_H_B], netB,
              netA, nullptr, EPI_ADD);

  // --- soft_agg(ii*12345+jj): netB -> netA ---
  launch_gemm(stream, netB, DIMC, nullptr, f_in[P_AIJ_G_W], f_in[P_AIJ_G_B], buf2,
              nullptr, nullptr, EPI_EXP);
  zero_kernel<<<2 * EW_BLOCKS, 256, 0, stream>>>(denom, 2 * SLOT);
  scatter_add_kernel<<<EW_BLOCKS, 256, 0, stream>>>(buf2, iij, denom);
  launch_gemm(stream, netB, DIMC, nullptr, f_in[P_AIJ_F_W], f_in[P_AIJ_F_B], buf1,
              nullptr, nullptr, EPI_NONE);
  weighted_kernel<<<EW_BLOCKS, 256, 0, stream>>>(buf2, denom, buf1, buf3);
  scatter_add_kernel<<<EW_BLOCKS, 256, 0, stream>>>(buf3, iij, ybuf);
  launch_gemm(stream, ybuf, DIMC, nullptr, f_in[P_AIJ_H_W], f_in[P_AIJ_H_B], netA,
              netB, nullptr, EPI_ADD);

  // --- LN + gated residual #1: netA -> netB(ln) -> netA ---
  layernorm_kernel<<<NTOK, 128, 0, stream>>>(netA, f_in[P_GRU_LN1_G], f_in[P_GRU_LN1_B],
                                             netB, 0);
  launch_gemm(stream, netB, DIMC, nullptr, f_in[P_GR1_GATE_W], f_in[P_GR1_GATE_B], buf2,
              nullptr, nullptr, EPI_SIGMOID);
  launch_gemm(stream, netB, DIMC, nullptr, f_in[P_GR1_RES1_W], f_in[P_GR1_RES1_B], buf1,
              nullptr, nullptr, EPI_RELU);
  launch_gemm(stream, buf1, DIMC, nullptr, f_in[P_GR1_RES2_W], f_in[P_GR1_RES2_B], netA,
              netB, buf2, EPI_GATED_ADD);

  // --- LN + gated residual #2: netA -> netB(ln) -> out_net ---
  layernorm_kernel<<<NTOK, 128, 0, stream>>>(netA, f_in[P_GRU_LN2_G], f_in[P_GRU_LN2_B],
                                             netB, 0);
  launch_gemm(stream, netB, DIMC, nullptr, f_in[P_GR2_GATE_W], f_in[P_GR2_GATE_B], buf2,
              nullptr, nullptr, EPI_SIGMOID);
  launch_gemm(stream, netB, DIMC, nullptr, f_in[P_GR2_RES1_W], f_in[P_GR2_RES1_B], buf1,
              nullptr, nullptr, EPI_RELU);
  launch_gemm(stream, buf1, DIMC, nullptr, f_in[P_GR2_RES2_W], f_in[P_GR2_RES2_B], out_net,
              netB, buf2, EPI_GATED_ADD);

  // --- heads ---
  heads_kernel<<<NTOK / 8, 256, 0, stream>>>(out_net, f_in[P_D_W], f_in[P_D_B],
                                             f_in[P_W_W], f_in[P_W_B], out_d, out_w);
}